// GRUBase_33268816675300
// MI455X (gfx1250) — compile-verified
//
#include <hip/hip_runtime.h>

// ---------------------------------------------------------------------------
// Bidirectional GRU (H=6, T=40) + FC(480 -> 10), B=65536, fp32 end-to-end.
//
// Strategy (MI455X / gfx1250, wave32):
//  * Recurrence is a serial chain of tiny 6x18 matvecs -> scalar fp32 VALU,
//    one thread per batch row (65536-way parallel, compute-bound: ~2 GFLOP
//    vs ~13 MB of HBM traffic => 23.3 TB/s makes memory a non-issue).
//  * Final FC is a real GEMM (B x 480) @ (480 x 10) -> V_WMMA_F32_16X16X4_F32
//    in fp32 so results match the reference exactly.
//  * x tile is staged memory->LDS with gfx1250 async copies (ASYNCcnt path),
//    no VGPR round-trip.
//  * LDS is the lane-transpose medium between per-thread features and the
//    WMMA A-fragment layout. Feature row stride 241 floats => conflict-free
//    16-row column gathers (gcd(241,64)=1).
// ---------------------------------------------------------------------------

typedef __attribute__((ext_vector_type(2))) float v2f;
typedef __attribute__((ext_vector_type(8))) float v8f;
typedef int v4i __attribute__((vector_size(4 * sizeof(int))));

#define AS1 __attribute__((address_space(1)))
#define AS3 __attribute__((address_space(3)))

#define TT    40      // timesteps
#define HH    6       // hidden size
#define G3    18      // 3*H gates
#define NCLS  10      // classes
#define KFULL 480     // 2*H*T
#define KHALF 240     // H*T (per direction)
#define ROWS  32      // batch rows per workgroup == 1 wave32
#define XSTR  44      // padded LDS stride for x rows (multiple of 4 -> 16B-aligned b128 chunks)
#define FSTR  241     // padded LDS stride for feature rows (odd -> conflict-free)

__device__ __forceinline__ float sigm_(float v) {
  return 1.0f / (1.0f + __expf(-v));
}
__device__ __forceinline__ float tanh_(float v) {
  // tanh(x) = 1 - 2/(exp(2x)+1); exact limits at +/-inf.
  return 1.0f - 2.0f / (__expf(2.0f * v) + 1.0f);
}

// One GRU direction for one batch row, fp32, h kept in VGPRs.
// sw layout: [0,18) w_ih | [18,36) b_ih | [36,144) w_hh (18x6 row-major) | [144,162) b_hh
__device__ __forceinline__ void gru_scan(const float* __restrict__ sw,
                                         const float* __restrict__ sxrow,
                                         float* __restrict__ frow,
                                         int bwd)
{
  const float* wih = sw;
  const float* bih = sw + 18;
  const float* whh = sw + 36;
  const float* bhh = sw + 144;

  float h[HH];
#pragma unroll
  for (int j = 0; j < HH; ++j) h[j] = 0.0f;

  for (int s = 0; s < TT; ++s) {
    const int t = bwd ? (TT - 1 - s) : s;
    const float xt = sxrow[t];

    float hp[G3];
#pragma unroll
    for (int g = 0; g < G3; ++g) {
      float acc = bhh[g];
#pragma unroll
      for (int j = 0; j < HH; ++j) acc = fmaf(whh[g * HH + j], h[j], acc);
      hp[g] = acc;
    }

    float hn[HH];
#pragma unroll
    for (int j = 0; j < HH; ++j) {
      const float xr = fmaf(xt, wih[j],      bih[j]);
      const float xz = fmaf(xt, wih[6 + j],  bih[6 + j]);
      const float xn = fmaf(xt, wih[12 + j], bih[12 + j]);
      const float r  = sigm_(xr + hp[j]);
      const float z  = sigm_(xz + hp[6 + j]);
      const float n  = tanh_(fmaf(r, hp[12 + j], xn));
      hn[j] = fmaf(z, h[j] - n, n);          // (1-z)*n + z*h
    }
#pragma unroll
    for (int j = 0; j < HH; ++j) {
      h[j] = hn[j];
      frow[t * HH + j] = hn[j];              // per-direction feature slot
    }
  }
}

// Accumulate one direction's FC contribution with fp32 WMMA.
// A (16x4 f32): lane l holds A[M=l%16][k0+2*(l>=16)+{0,1}]
// B (4x16 f32): lane l holds B[k0+2*(l>=16)+{0,1}][N=l%16]
// C/D (16x16 f32): VGPR v, lane l -> D[v + 8*(l>=16)][l%16]
// Feature k maps to flat column (k/6)*12 + (k%6) + colOff (colOff: 0 fwd, 6 bwd).
__device__ __forceinline__ void fc_accum(const float* __restrict__ s_feat,
                                         const float* __restrict__ fc_w,
                                         int laneM, int laneHalf, int colOff,
                                         v8f& c0, v8f& c1)
{
  const int nClamped = (laneM < NCLS) ? laneM : (NCLS - 1);
  const float nMask = (laneM < NCLS) ? 1.0f : 0.0f;  // no divergence around WMMA
  const float* fw = fc_w + nClamped * KFULL + colOff;

#pragma unroll 4
  for (int c = 0; c < KHALF / 4; ++c) {
    const int k0 = 4 * c + 2 * laneHalf;
    const int k1 = k0 + 1;

    v2f bf;
    bf.x = nMask * fw[(k0 / HH) * 12 + (k0 % HH)];
    bf.y = nMask * fw[(k1 / HH) * 12 + (k1 % HH)];

    v2f a0, a1;
    a0.x = s_feat[laneM * FSTR + k0];
    a0.y = s_feat[laneM * FSTR + k1];
    a1.x = s_feat[(16 + laneM) * FSTR + k0];
    a1.y = s_feat[(16 + laneM) * FSTR + k1];

    c0 = __builtin_amdgcn_wmma_f32_16x16x4_f32(false, a0, false, bf,
                                               (short)0, c0, false, false);
    c1 = __builtin_amdgcn_wmma_f32_16x16x4_f32(false, a1, false, bf,
                                               (short)0, c1, false, false);
  }
}

__global__ __launch_bounds__(ROWS)
void gru_bidir_fc_kernel(const float* __restrict__ x,
                         const float* __restrict__ w_ih_f, const float* __restrict__ w_hh_f,
                         const float* __restrict__ b_ih_f, const float* __restrict__ b_hh_f,
                         const float* __restrict__ w_ih_b, const float* __restrict__ w_hh_b,
                         const float* __restrict__ b_ih_b, const float* __restrict__ b_hh_b,
                         const float* __restrict__ fc_w,  const float* __restrict__ fc_b,
                         float* __restrict__ out, int batch)
{
  __shared__ float s_x[ROWS * XSTR];       //  5,632 B
  __shared__ float s_feat[ROWS * FSTR];    // 30,848 B (reused fwd then bwd)
  __shared__ float s_wf[162];              //    648 B
  __shared__ float s_wb[162];              //    648 B

  const int tid      = threadIdx.x;        // 0..31
  const int rowBase  = blockIdx.x * ROWS;
  const int laneM    = tid & 15;
  const int laneHalf = tid >> 4;           // 0: K={0,1}, 1: K={2,3} halves

  // Stage the tiny weights into LDS (broadcast reads in the scan loops).
  if (tid < G3) {
    s_wf[tid] = w_ih_f[tid];  s_wf[18 + tid] = b_ih_f[tid];  s_wf[144 + tid] = b_hh_f[tid];
    s_wb[tid] = w_ih_b[tid];  s_wb[18 + tid] = b_ih_b[tid];  s_wb[144 + tid] = b_hh_b[tid];
  }
  for (int i = tid; i < G3 * HH; i += ROWS) {
    s_wf[36 + i] = w_hh_f[i];
    s_wb[36 + i] = w_hh_b[i];
  }

  const int xlim = batch * TT;

#if __has_builtin(__builtin_amdgcn_global_load_async_to_lds_b128)
  // gfx1250 async memory->LDS copy: 16B per lane per op, ASYNCcnt-tracked.
  // 32 rows x 40 floats = 320 chunks of 4 floats -> 10 wave-wide b128 ops.
  {
    const float* xblk = x + (size_t)rowBase * TT;
#pragma unroll
    for (int it = 0; it < (ROWS * TT) / (32 * 4); ++it) {   // 10 iterations
      const int cid = it * 32 + tid;        // chunk id 0..319
      const int r   = cid / 10;             // row within tile
      const int c   = (cid - r * 10) * 4;   // col (multiple of 4 -> 16B aligned)
      const int gi  = rowBase * TT + r * TT + c;
      if (gi + 3 < xlim) {
        __builtin_amdgcn_global_load_async_to_lds_b128(
            (AS1 v4i*)(xblk + r * TT + c),
            (AS3 v4i*)(&s_x[r * XSTR + c]),
            /*imm offset=*/0, /*cpol=*/0);
      }
    }
  }
#if __has_builtin(__builtin_amdgcn_s_wait_asynccnt)
  __builtin_amdgcn_s_wait_asynccnt(0);
#else
  asm volatile("s_wait_asynccnt 0x0" ::: "memory");
#endif
#else
  // Fallback: VGPR round-trip staging.
  for (int i = tid; i < ROWS * TT; i += ROWS) {
    const int gi = rowBase * TT + i;
    s_x[(i / TT) * XSTR + (i % TT)] = (gi < xlim) ? x[gi] : 0.0f;
  }
#endif
  __syncthreads();

  v8f c0 = {0.0f, 0.0f, 0.0f, 0.0f, 0.0f, 0.0f, 0.0f, 0.0f};
  v8f c1 = {0.0f, 0.0f, 0.0f, 0.0f, 0.0f, 0.0f, 0.0f, 0.0f};

  // Forward direction: scan, then WMMA-accumulate its FC contribution.
  gru_scan(s_wf, s_x + tid * XSTR, s_feat + tid * FSTR, /*bwd=*/0);
  __syncthreads();
  fc_accum(s_feat, fc_w, laneM, laneHalf, /*colOff=*/0, c0, c1);
  __syncthreads();

  // Backward direction reuses the same LDS feature tile.
  gru_scan(s_wb, s_x + tid * XSTR, s_feat + tid * FSTR, /*bwd=*/1);
  __syncthreads();
  fc_accum(s_feat, fc_w, laneM, laneHalf, /*colOff=*/HH, c0, c1);

  // Scatter D fragments: row = base + 8*laneHalf + v, col = laneM (< NCLS).
  if (laneM < NCLS) {
    const float bias = fc_b[laneM];
#pragma unroll
    for (int v = 0; v < 8; ++v) {
      const int r0 = rowBase + laneHalf * 8 + v;
      if (r0 < batch) out[r0 * NCLS + laneM] = c0[v] + bias;
      const int r1 = rowBase + 16 + laneHalf * 8 + v;
      if (r1 < batch) out[r1 * NCLS + laneM] = c1[v] + bias;
    }
  }
}

extern "C" void kernel_launch(void* const* d_in, const int* in_sizes, int n_in,
                              void* d_out, int out_size, void* d_ws, size_t ws_size,
                              hipStream_t stream) {
  const float* x      = (const float*)d_in[0];
  const float* w_ih_f = (const float*)d_in[1];
  const float* w_hh_f = (const float*)d_in[2];
  const float* b_ih_f = (const float*)d_in[3];
  const float* b_hh_f = (const float*)d_in[4];
  const float* w_ih_b = (const float*)d_in[5];
  const float* w_hh_b = (const float*)d_in[6];
  const float* b_ih_b = (const float*)d_in[7];
  const float* b_hh_b = (const float*)d_in[8];
  const float* fc_w   = (const float*)d_in[9];
  const float* fc_b   = (const float*)d_in[10];
  float* out = (float*)d_out;

  const int batch = in_sizes[0] / TT;
  const int grid  = (batch + ROWS - 1) / ROWS;

  hipLaunchKernelGGL(gru_bidir_fc_kernel, dim3(grid), dim3(ROWS), 0, stream,
                     x, w_ih_f, w_hh_f, b_ih_f, b_hh_f,
                     w_ih_b, w_hh_b, b_ih_b, b_hh_b,
                     fc_w, fc_b, out, batch);
  (void)d_ws; (void)ws_size; (void)n_in; (void)out_size;
}